// KeypointLoss_73194832659123
// MI455X (gfx1250) — compile-verified
//
#include <hip/hip_runtime.h>
#include <math.h>

// Problem constants (from reference): S=NSTACK+1=3, B=8, K=11, C=2K+1=23, H=W=256
#define S_DIM 3
#define B_DIM 8
#define K_DIM 11
#define C_DIM 23
#define H_DIM 256
#define HW    65536

typedef float v2f __attribute__((ext_vector_type(2)));
typedef float v4f __attribute__((ext_vector_type(4)));
typedef float v8f __attribute__((ext_vector_type(8)));

// ---- wave32 shuffle reduction -------------------------------------------
__device__ __forceinline__ float wave_reduce_add(float v) {
    v += __shfl_xor(v, 16, 32);
    v += __shfl_xor(v, 8, 32);
    v += __shfl_xor(v, 4, 32);
    v += __shfl_xor(v, 2, 32);
    v += __shfl_xor(v, 1, 32);
    return v;
}

__device__ __forceinline__ float bce_elem(float p, float g) {
    float lp  = fmaxf(__logf(p), -100.0f);
    float l1p = fmaxf(log1pf(-p), -100.0f);
    return -(g * lp + (1.0f - g) * l1p);
}

// ws layout (floats): [0..23] heat sums, [24..47] label sums, [48..71] mask sums
__global__ void kl_init(float* __restrict__ ws) {
    if (threadIdx.x < 72) ws[threadIdx.x] = 0.0f;
}

// ---- heat loss: sum((hm_pred*mask - heat)^2) over HW per (s,b,k) --------
// WMMA f32 16x16x4 with B = ones acts as an *exact* f32 summation engine:
// D[m,n] += sum_k A[m,k].  Every row-sum is replicated across 16 columns,
// so (sum of all 256 accumulator slots)/16 == exact sum of all fed values.
__global__ void kl_heat(const float* __restrict__ cp,
                        const float* __restrict__ hm,
                        const float* __restrict__ mk,
                        float* __restrict__ ws) {
    const int sbk = blockIdx.x;                 // 0..S*B*K-1
    const int s   = sbk / (B_DIM * K_DIM);
    const int rem = sbk % (B_DIM * K_DIM);
    const int b   = rem / K_DIM;
    const int k   = rem % K_DIM;
    const int sb  = s * B_DIM + b;

    const float* __restrict__ hp = cp + (size_t)(sb * C_DIM + K_DIM + k) * HW; // hm_pred channel
    const float* __restrict__ hh = hm + (size_t)(sb * K_DIM + k) * HW;
    const float* __restrict__ mm = mk + (size_t)sb * HW;

    const int base = blockIdx.y * (HW / 8) + (int)threadIdx.x * 4;  // 8 chunks of 8192

    v8f acc = {};                      // 16x16 f32 accumulator (8 VGPRs/lane)
    const v2f ones = {1.0f, 1.0f};

#pragma unroll
    for (int it = 0; it < 8; ++it) {
        const int p = base + it * 1024;        // 256 thr * 4 floats per iter
        // hm_pred is stream-once -> non-temporal; heatmaps/masks are reused
        // by later kernels -> regular temporal so they stay in the 192MB L2.
        v4f xp = __builtin_nontemporal_load((const v4f*)(hp + p));
        v4f xh = *(const v4f*)(hh + p);
        v4f xm = *(const v4f*)(mm + p);
        __builtin_prefetch(hp + p + 1024, 0, 1);

        v4f d;
        d.x = fmaf(xp.x, xm.x, -xh.x);
        d.y = fmaf(xp.y, xm.y, -xh.y);
        d.z = fmaf(xp.z, xm.z, -xh.z);
        d.w = fmaf(xp.w, xm.w, -xh.w);

        v2f a0 = {d.x * d.x, d.y * d.y};
        v2f a1 = {d.z * d.z, d.w * d.w};
        acc = __builtin_amdgcn_wmma_f32_16x16x4_f32(false, a0, false, ones,
                                                    (short)0, acc, false, false);
        acc = __builtin_amdgcn_wmma_f32_16x16x4_f32(false, a1, false, ones,
                                                    (short)0, acc, false, false);
    }

    float lane = acc[0] + acc[1] + acc[2] + acc[3] + acc[4] + acc[5] + acc[6] + acc[7];
    float tot  = wave_reduce_add(lane) * (1.0f / 16.0f);  // undo 16x column replication
    if ((threadIdx.x & 31) == 0) atomicAdd(&ws[sb], tot);
}

// ---- argmax + label BCE per (s,b,k) -------------------------------------
__global__ void kl_label(const float* __restrict__ cp,
                         const float* __restrict__ hm,
                         const float* __restrict__ lbl,
                         float* __restrict__ ws) {
    const int sbk = blockIdx.x;
    const int s   = sbk / (B_DIM * K_DIM);
    const int rem = sbk % (B_DIM * K_DIM);
    const int b   = rem / K_DIM;
    const int k   = rem % K_DIM;
    const int sb  = s * B_DIM + b;

    const float* __restrict__ hh = hm + (size_t)(sb * K_DIM + k) * HW;

    float best = -1.0f;
    int   bidx = 0x7fffffff;
#pragma unroll 4
    for (int j = 0; j < HW / 256; ++j) {
        const int p = (int)threadIdx.x + j * 256;      // coalesced per iteration
        float v = __builtin_nontemporal_load(hh + p);  // last use of heatmaps
        if (v > best) { best = v; bidx = p; }          // strided order keeps
        else if (v == best && p < bidx) { bidx = p; }  // first-index tie-break
    }

    __shared__ float sval[256];
    __shared__ int   sidx[256];
    sval[threadIdx.x] = best;
    sidx[threadIdx.x] = bidx;
    __syncthreads();
#pragma unroll
    for (int off = 128; off > 0; off >>= 1) {
        if ((int)threadIdx.x < off) {
            float v1 = sval[threadIdx.x],       v2 = sval[threadIdx.x + off];
            int   i1 = sidx[threadIdx.x],       i2 = sidx[threadIdx.x + off];
            if (v2 > v1 || (v2 == v1 && i2 < i1)) {
                sval[threadIdx.x] = v2;
                sidx[threadIdx.x] = i2;
            }
        }
        __syncthreads();
    }

    if (threadIdx.x == 0) {
        const float peak = sval[0];
        const int   idx  = sidx[0];          // H==W  =>  lin == idx
        float contrib = 0.0f;
        if (peak == 1.0f) {                  // exact: setup writes literal 1.0f
            float bs = 0.0f;
#pragma unroll
            for (int c = 0; c < 7; ++c) {
                float p = cp[(size_t)(sb * C_DIM + c) * HW + idx];
                float g = lbl[(b * K_DIM + k) * 7 + c];
                bs += bce_elem(p, g);
            }
            contrib = bs * (1.0f / 7.0f);
        }
        atomicAdd(&ws[24 + sb], contrib);
    }
}

// ---- mask BCE per (s,b) --------------------------------------------------
__global__ void kl_mask(const float* __restrict__ cp,
                        const float* __restrict__ mk,
                        float* __restrict__ ws) {
    const int sb = blockIdx.x;                                  // 0..23
    const float* __restrict__ pp = cp + (size_t)(sb * C_DIM + 2 * K_DIM) * HW;
    const float* __restrict__ mm = mk + (size_t)sb * HW;

    const int base = blockIdx.y * (HW / 16) + (int)threadIdx.x * 4;  // 16 chunks

    float acc = 0.0f;
#pragma unroll
    for (int it = 0; it < 4; ++it) {
        const int p = base + it * 1024;
        v4f xp = __builtin_nontemporal_load((const v4f*)(pp + p)); // stream-once
        v4f xg = *(const v4f*)(mm + p);                            // L2-resident
        acc += bce_elem(xp.x, xg.x);
        acc += bce_elem(xp.y, xg.y);
        acc += bce_elem(xp.z, xg.z);
        acc += bce_elem(xp.w, xg.w);
    }
    float tot = wave_reduce_add(acc);
    if ((threadIdx.x & 31) == 0) atomicAdd(&ws[48 + sb], tot);
}

// ---- finalize: scale + transpose (S,B)->(B,S), write 72 outputs ---------
__global__ void kl_finalize(const float* __restrict__ ws, float* __restrict__ out) {
    const int t = threadIdx.x;
    if (t < S_DIM * B_DIM) {
        const int b  = t / S_DIM;
        const int s  = t % S_DIM;
        const int sb = s * B_DIM + b;
        out[t]      = ws[sb]      * (1.0f / K_DIM);        // heat_l.T
        out[24 + t] = ws[24 + sb] * (1.0f / K_DIM);        // label_l.T
        out[48 + t] = ws[48 + sb] * (1.0f / (float)HW);    // mask_l.T
    }
}

extern "C" void kernel_launch(void* const* d_in, const int* in_sizes, int n_in,
                              void* d_out, int out_size, void* d_ws, size_t ws_size,
                              hipStream_t stream) {
    const float* cp  = (const float*)d_in[0];  // combined_preds (S,B,C,H,W)
    const float* hm  = (const float*)d_in[1];  // heatmaps       (S,B,K,H,W)
    const float* lbl = (const float*)d_in[2];  // labels         (B,K,7)
    const float* mk  = (const float*)d_in[3];  // masks          (S,B,1,H,W)
    float* ws  = (float*)d_ws;
    float* out = (float*)d_out;

    kl_init<<<1, 128, 0, stream>>>(ws);
    // heat first: pulls heatmaps + masks into L2 with RT hints for reuse below
    kl_heat<<<dim3(S_DIM * B_DIM * K_DIM, 8), 256, 0, stream>>>(cp, hm, mk, ws);
    kl_label<<<S_DIM * B_DIM * K_DIM, 256, 0, stream>>>(cp, hm, lbl, ws);
    kl_mask<<<dim3(S_DIM * B_DIM, 16), 256, 0, stream>>>(cp, mk, ws);
    kl_finalize<<<1, 32, 0, stream>>>(ws, out);
}